// ANet_sm_ts_57191784513796
// MI455X (gfx1250) — compile-verified
//
#include <hip/hip_runtime.h>
#include <hip/hip_bf16.h>

typedef float v2f __attribute__((ext_vector_type(2)));
typedef float v8f __attribute__((ext_vector_type(8)));

__device__ __forceinline__ int clampi(int v, int lo, int hi) {
    return v < lo ? lo : (v > hi ? hi : v);
}

// ---------------------------------------------------------------------------
// Specialized conv1x1 GEMM: Y[m,s] = sum_k W[m,k] * X[k,s], M = 36 fixed,
// K compile-time (1 or 36).  A-fragments (weights) preloaded into registers
// once per wave and reused across a chunk of 4 N-tiles.  All edge handling is
// clamp+select (branch-free) so the K-loop is pure loads + v_wmma.
// ---------------------------------------------------------------------------
template <int K>
__global__ __launch_bounds__(256) void conv1x1_wmma_kernel(
    const float* __restrict__ W, const float* __restrict__ X,
    float* __restrict__ Y, int S)
{
    const int M = 36;
    constexpr int KT = (K + 3) / 4;
    int ntiles  = (S + 15) >> 4;
    int nchunks = (ntiles + 3) >> 2;
    int chunks  = 3 * nchunks;                       // mtiles = 3
    int wave = blockIdx.x * (blockDim.x >> 5) + (threadIdx.x >> 5);
    if (wave >= chunks) return;                      // wave-uniform
    int mt = wave / nchunks;
    int nc = wave % nchunks;
    int lane = threadIdx.x & 31;
    int half = lane >> 4;
    int l16  = lane & 15;
    int m0 = mt << 4;
    int ar  = m0 + l16;
    int arc = clampi(ar, 0, M - 1);                  // clamped row (always valid)
    bool arok = (ar < M);

    // Preload A fragments for all K-steps (branch-free masked loads)
    v2f af[KT];
#pragma unroll
    for (int kt = 0; kt < KT; ++kt) {
        int ac0 = (kt << 2) + (half << 1);
        int ac1 = ac0 + 1;
        float w0 = W[(long)arc * K + clampi(ac0, 0, K - 1)];
        float w1 = W[(long)arc * K + clampi(ac1, 0, K - 1)];
        af[kt].x = (arok && ac0 < K) ? w0 : 0.f;
        af[kt].y = (arok && ac1 < K) ? w1 : 0.f;
    }

#pragma unroll
    for (int t = 0; t < 4; ++t) {
        int nt = (nc << 2) + t;
        if (nt >= ntiles) break;                     // wave-uniform
        int bc  = (nt << 4) + l16;
        int bcc = clampi(bc, 0, S - 1);
        bool bcok = (bc < S);
        // prefetch next tile's B column (gfx1250 global_prefetch path)
        __builtin_prefetch(X + clampi(bc + 16, 0, S - 1), 0, 1);
        v8f acc = {0.f, 0.f, 0.f, 0.f, 0.f, 0.f, 0.f, 0.f};
#pragma unroll
        for (int kt = 0; kt < KT; ++kt) {
            int ac0 = (kt << 2) + (half << 1);
            int ac1 = ac0 + 1;
            float x0 = X[(long)clampi(ac0, 0, K - 1) * S + bcc];
            float x1 = X[(long)clampi(ac1, 0, K - 1) * S + bcc];
            v2f b;
            b.x = (bcok && ac0 < K) ? x0 : 0.f;
            b.y = (bcok && ac1 < K) ? x1 : 0.f;
            acc = __builtin_amdgcn_wmma_f32_16x16x4_f32(
                false, af[kt], false, b, (short)0, acc, false, false);
        }
#pragma unroll
        for (int i = 0; i < 8; ++i) {
            int r = m0 + (half << 3) + i;
            if (r < M && bcok) Y[(long)r * S + bc] = acc[i];
        }
    }
}

// ---------------------------------------------------------------------------
// Linear (GEMV) via WMMA: y[m] = act( W[m,:] . x + bias[m] ).  S==1 column.
// Requires K % 4 == 0 (true for all linear layers here) -> no K bounds checks.
// ---------------------------------------------------------------------------
__global__ __launch_bounds__(256) void linear_wmma_kernel(
    const float* __restrict__ W, const float* __restrict__ x,
    const float* __restrict__ bias, float* __restrict__ y,
    int M, int K, int relu)
{
    int mtiles = (M + 15) >> 4;
    int wave = blockIdx.x * (blockDim.x >> 5) + (threadIdx.x >> 5);
    if (wave >= mtiles) return;
    int lane = threadIdx.x & 31;
    int half = lane >> 4;
    int l16  = lane & 15;
    int m0 = wave << 4;
    int ar  = m0 + l16;
    int arc = clampi(ar, 0, M - 1);
    bool arok  = (ar < M);
    bool col0  = (l16 == 0);
    v8f acc = {0.f, 0.f, 0.f, 0.f, 0.f, 0.f, 0.f, 0.f};
#pragma unroll 4
    for (int k0 = 0; k0 < K; k0 += 4) {
        int ac = k0 + (half << 1);
        float w0 = W[(long)arc * K + ac];
        float w1 = W[(long)arc * K + ac + 1];
        float x0 = x[ac];
        float x1 = x[ac + 1];
        v2f a, b;
        a.x = arok ? w0 : 0.f;
        a.y = arok ? w1 : 0.f;
        b.x = col0 ? x0 : 0.f;
        b.y = col0 ? x1 : 0.f;
        acc = __builtin_amdgcn_wmma_f32_16x16x4_f32(
            false, a, false, b, (short)0, acc, false, false);
    }
#pragma unroll
    for (int i = 0; i < 8; ++i) {
        int r = m0 + (half << 3) + i;
        if (r < M && col0) {
            float v = acc[i];
            if (bias) v += bias[r];
            if (relu) v = fmaxf(v, 0.f);
            y[r] = v;
        }
    }
}

// ---------------------------------------------------------------------------
// Build 3D relative-position table rel[36][k^3] from rel_d/h/w [12][k]
// ---------------------------------------------------------------------------
__global__ void build_rel3d_kernel(const float* __restrict__ rd,
                                   const float* __restrict__ rh,
                                   const float* __restrict__ rw,
                                   float* __restrict__ rel, int ksz)
{
    int k3 = ksz * ksz * ksz;
    int idx = blockIdx.x * blockDim.x + threadIdx.x;
    if (idx >= 36 * k3) return;
    int w = idx % k3;
    int c = idx / k3;
    int i = w / (ksz * ksz);
    int j = (w / ksz) % ksz;
    int l = w % ksz;
    float v;
    if (c < 12)      v = rd[c * ksz + i];
    else if (c < 24) v = rh[(c - 12) * ksz + j];
    else             v = rw[(c - 24) * ksz + l];
    rel[idx] = v;
}

// ---------------------------------------------------------------------------
// 3D local-attention conv, branch-free (clamped) window loads.
// ---------------------------------------------------------------------------
__global__ void attn3d_kernel(const float* __restrict__ q,
                              const float* __restrict__ kx,
                              const float* __restrict__ vx,
                              const float* __restrict__ rel,
                              float* __restrict__ out,
                              int Din, int Dout, int ksz, int pad)
{
    const int C = 36;
    long idx = (long)blockIdx.x * blockDim.x + threadIdx.x;
    long total = (long)C * Dout * Dout * Dout;
    if (idx >= total) return;
    int xo = (int)(idx % Dout); long t = idx / Dout;
    int yo = (int)(t % Dout);   t /= Dout;
    int zo = (int)(t % Dout);
    int c  = (int)(t / Dout);
    long DD = (long)Din * Din;
    const float* qc = q  + (long)c * Din * DD;
    const float* kc = kx + (long)c * Din * DD;
    const float* vc = vx + (long)c * Din * DD;
    const float* rc = rel + (long)c * ksz * ksz * ksz;
    float qv = qc[(long)zo * DD + (long)yo * Din + xo];
    // pass 1: window max
    float mx = -3.4e38f;
    int w = 0;
    for (int i = 0; i < ksz; ++i) {
        int zi = zo + i - pad;
        int zc = clampi(zi, 0, Din - 1);
        bool zok = (unsigned)zi < (unsigned)Din;
        for (int j = 0; j < ksz; ++j) {
            int yi = yo + j - pad;
            int yc = clampi(yi, 0, Din - 1);
            bool yok = (unsigned)yi < (unsigned)Din;
            for (int l = 0; l < ksz; ++l, ++w) {
                int xi = xo + l - pad;
                int xc = clampi(xi, 0, Din - 1);
                bool inr = zok && yok && ((unsigned)xi < (unsigned)Din);
                float kl = kc[(long)zc * DD + (long)yc * Din + xc];
                float kv = rc[w] + (inr ? kl : 0.f);
                mx = fmaxf(mx, qv * kv);
            }
        }
    }
    // pass 2: exp / sum / weighted v
    float sum = 0.f, accv = 0.f;
    w = 0;
    for (int i = 0; i < ksz; ++i) {
        int zi = zo + i - pad;
        int zc = clampi(zi, 0, Din - 1);
        bool zok = (unsigned)zi < (unsigned)Din;
        for (int j = 0; j < ksz; ++j) {
            int yi = yo + j - pad;
            int yc = clampi(yi, 0, Din - 1);
            bool yok = (unsigned)yi < (unsigned)Din;
            for (int l = 0; l < ksz; ++l, ++w) {
                int xi = xo + l - pad;
                int xc = clampi(xi, 0, Din - 1);
                bool inr = zok && yok && ((unsigned)xi < (unsigned)Din);
                long off = (long)zc * DD + (long)yc * Din + xc;
                float kl = kc[off];
                float vl = vc[off];
                float kv = rc[w] + (inr ? kl : 0.f);
                float e = __expf(qv * kv - mx);
                sum += e;
                accv += e * (inr ? vl : 0.f);
            }
        }
    }
    out[idx] = accv / sum;
}

// ---------------------------------------------------------------------------
// 1D local-attention conv, branch-free window loads.
// ---------------------------------------------------------------------------
__global__ void attn1d_kernel(const float* __restrict__ q,
                              const float* __restrict__ kx,
                              const float* __restrict__ vx,
                              const float* __restrict__ rel,
                              float* __restrict__ out,
                              int Lin, int Lout, int ksz, int pad)
{
    const int C = 36;
    long idx = (long)blockIdx.x * blockDim.x + threadIdx.x;
    long total = (long)C * Lout;
    if (idx >= total) return;
    int s = (int)(idx % Lout);
    int c = (int)(idx / Lout);
    const float* kc = kx + (long)c * Lin;
    const float* vc = vx + (long)c * Lin;
    const float* rc = rel + c * ksz;
    float qv = q[(long)c * Lin + s];
    float mx = -3.4e38f;
    for (int i = 0; i < ksz; ++i) {
        int si = s + i - pad;
        int sc = clampi(si, 0, Lin - 1);
        bool inr = (unsigned)si < (unsigned)Lin;
        float kl = kc[sc];
        float kv = rc[i] + (inr ? kl : 0.f);
        mx = fmaxf(mx, qv * kv);
    }
    float sum = 0.f, accv = 0.f;
    for (int i = 0; i < ksz; ++i) {
        int si = s + i - pad;
        int sc = clampi(si, 0, Lin - 1);
        bool inr = (unsigned)si < (unsigned)Lin;
        float kl = kc[sc];
        float vl = vc[sc];
        float kv = rc[i] + (inr ? kl : 0.f);
        float e = __expf(qv * kv - mx);
        sum += e;
        accv += e * (inr ? vl : 0.f);
    }
    out[idx] = accv / sum;
}

// ---------------------------------------------------------------------------
// 3D max-pool (PxPxP, stride P) with optional input-ReLU and accumulate
// ---------------------------------------------------------------------------
__global__ void pool3d_kernel(const float* __restrict__ in, float* __restrict__ out,
                              int Din, int Psz, int relu, int accum)
{
    const int C = 36;
    int Dout = Din / Psz;
    long idx = (long)blockIdx.x * blockDim.x + threadIdx.x;
    long total = (long)C * Dout * Dout * Dout;
    if (idx >= total) return;
    int xo = (int)(idx % Dout); long t = idx / Dout;
    int yo = (int)(t % Dout);   t /= Dout;
    int zo = (int)(t % Dout);
    int c  = (int)(t / Dout);
    long DD = (long)Din * Din;
    const float* base = in + (long)c * Din * DD;
    float m = -3.4e38f;
    for (int i = 0; i < Psz; ++i)
        for (int j = 0; j < Psz; ++j)
            for (int l = 0; l < Psz; ++l) {
                float v = base[(long)(zo * Psz + i) * DD +
                               (long)(yo * Psz + j) * Din + (xo * Psz + l)];
                m = fmaxf(m, v);
            }
    if (relu) m = fmaxf(m, 0.f);
    if (accum) out[idx] += m; else out[idx] = m;
}

__global__ void pool1d_kernel(const float* __restrict__ in, float* __restrict__ out,
                              int Lin, int Psz, int relu, int accum)
{
    const int C = 36;
    int Lout = Lin / Psz;
    long idx = (long)blockIdx.x * blockDim.x + threadIdx.x;
    long total = (long)C * Lout;
    if (idx >= total) return;
    int so = (int)(idx % Lout);
    int c  = (int)(idx / Lout);
    const float* base = in + (long)c * Lin;
    float m = -3.4e38f;
    for (int i = 0; i < Psz; ++i) m = fmaxf(m, base[so * Psz + i]);
    if (relu) m = fmaxf(m, 0.f);
    if (accum) out[idx] += m; else out[idx] = m;
}

// ---------------------------------------------------------------------------
// Channel attention: s = sigmoid(mlp(mean_sp) + mlp(max_sp)), x *= s in place
// ---------------------------------------------------------------------------
__global__ void channel_attn_kernel(float* __restrict__ x,
                                    const float* __restrict__ w1,
                                    const float* __restrict__ b1,
                                    const float* __restrict__ w2,
                                    const float* __restrict__ b2, int S)
{
    const int C = 36, H = 9;
    __shared__ float mean_s[C], max_s[C], hmean[H], hmax[H], scale[C];
    int t = threadIdx.x;
    if (t < C) {
        const float* xc = x + (long)t * S;
        float su = 0.f, m = -3.4e38f;
        for (int i = 0; i < S; ++i) { float v = xc[i]; su += v; m = fmaxf(m, v); }
        mean_s[t] = su / (float)S;
        max_s[t]  = m;
    }
    __syncthreads();
    if (t < H) {
        float a = b1[t], b = b1[t];
        for (int c = 0; c < C; ++c) {
            float w = w1[t * C + c];
            a += w * mean_s[c];
            b += w * max_s[c];
        }
        hmean[t] = fmaxf(a, 0.f);
        hmax[t]  = fmaxf(b, 0.f);
    }
    __syncthreads();
    if (t < C) {
        float a = b2[t], b = b2[t];
        for (int j = 0; j < H; ++j) {
            float w = w2[t * H + j];
            a += w * hmean[j];
            b += w * hmax[j];
        }
        float s = a + b;
        scale[t] = 1.f / (1.f + __expf(-s));
    }
    __syncthreads();
    for (long i = t; i < (long)C * S; i += blockDim.x) x[i] *= scale[i / S];
}

// ---------------------------------------------------------------------------
// Spatial attention (3D): channel mean/max -> m[2,S], then 7^3 conv + sigmoid
// ---------------------------------------------------------------------------
__global__ void chan_meanmax_kernel(const float* __restrict__ x,
                                    float* __restrict__ m, int S)
{
    const int C = 36;
    int s = blockIdx.x * blockDim.x + threadIdx.x;
    if (s >= S) return;
    float su = 0.f, mx = -3.4e38f;
    for (int c = 0; c < C; ++c) {
        float v = x[(long)c * S + s];
        su += v; mx = fmaxf(mx, v);
    }
    m[s]     = su / (float)C;
    m[S + s] = mx;
}

__global__ void sattn3d_kernel(float* __restrict__ x, const float* __restrict__ m,
                               const float* __restrict__ w, const float* __restrict__ b,
                               int D)
{
    int S = D * D * D;
    int idx = blockIdx.x * blockDim.x + threadIdx.x;
    if (idx >= S) return;
    int xo = idx % D;
    int yo = (idx / D) % D;
    int zo = idx / (D * D);
    float accv = b[0];
    for (int ch = 0; ch < 2; ++ch)
        for (int i = 0; i < 7; ++i) {
            int zi = zo + i - 3;
            bool zok = (unsigned)zi < (unsigned)D;
            int zc = clampi(zi, 0, D - 1);
            for (int j = 0; j < 7; ++j) {
                int yi = yo + j - 3;
                bool yok = zok && ((unsigned)yi < (unsigned)D);
                int yc = clampi(yi, 0, D - 1);
                for (int l = 0; l < 7; ++l) {
                    int xi = xo + l - 3;
                    bool inr = yok && ((unsigned)xi < (unsigned)D);
                    int xc = clampi(xi, 0, D - 1);
                    float mv = m[ch * S + (zc * D + yc) * D + xc];
                    accv += w[((ch * 7 + i) * 7 + j) * 7 + l] * (inr ? mv : 0.f);
                }
            }
        }
    float s = 1.f / (1.f + __expf(-accv));
    for (int c = 0; c < 36; ++c) x[(long)c * S + idx] *= s;
}

// ---------------------------------------------------------------------------
// Host orchestration
// ---------------------------------------------------------------------------
extern "C" void kernel_launch(void* const* d_in, const int* in_sizes, int n_in,
                              void* d_out, int out_size, void* d_ws, size_t ws_size,
                              hipStream_t stream)
{
    (void)in_sizes; (void)n_in; (void)out_size; (void)ws_size;
    auto P = [&](int i) -> const float* { return (const float*)d_in[i]; };

    // Workspace layout (floats)
    float* ws  = (float*)d_ws;
    float* Q   = ws;                 // 1,316,000 (max 36*35937)
    float* Kb  = Q  + 1316000;
    float* Vb  = Kb + 1316000;
    float* A   = Vb + 1316000;       // attn output, max 36*32768
    float* XA  = A  + 1200000;       // 160,000 each
    float* XB  = XA + 160000;
    float* XC  = XB + 160000;
    float* XD  = XC + 160000;
    float* REL = XD + 160000;        // 4096
    float* MM  = REL + 4096;         // 1024
    float* F1  = MM  + 1024;         // 2048
    float* Z   = F1  + 2048;         // 1024 (concat 832)
    float* H4  = Z   + 1024;         // 512
    float* H5  = H4  + 512;          // 256

    auto conv36 = [&](const float* W, const float* X, float* Y, int S) {
        int ntiles = (S + 15) / 16;
        int chunks = 3 * ((ntiles + 3) / 4);
        conv1x1_wmma_kernel<36><<<(chunks + 7) / 8, 256, 0, stream>>>(W, X, Y, S);
    };
    auto conv1 = [&](const float* W, const float* X, float* Y, int S) {
        int ntiles = (S + 15) / 16;
        int chunks = 3 * ((ntiles + 3) / 4);
        conv1x1_wmma_kernel<1><<<(chunks + 7) / 8, 256, 0, stream>>>(W, X, Y, S);
    };
    auto linear = [&](const float* W, const float* x, const float* bias, float* y,
                      int M, int K, int relu) {
        int mtiles = (M + 15) / 16;
        linear_wmma_kernel<<<(mtiles + 7) / 8, 256, 0, stream>>>(W, x, bias, y, M, K, relu);
    };
    auto launch1 = [&](long n) { return (int)((n + 255) / 256); };

    // =========================== SM branch (3D) ===========================
    {   // c1: Cin=1, Din=33, k=4, pad=1 -> Dout=32, pool2 -> XA [36,16^3]
        int Din = 33, Dout = 32, S = Din * Din * Din;
        conv1(P(2), P(0), Q,  S);
        conv1(P(3), P(0), Kb, S);
        conv1(P(4), P(0), Vb, S);
        build_rel3d_kernel<<<launch1(36 * 64), 256, 0, stream>>>(P(5), P(6), P(7), REL, 4);
        long ta = 36L * Dout * Dout * Dout;
        attn3d_kernel<<<launch1(ta), 256, 0, stream>>>(Q, Kb, Vb, REL, A, Din, Dout, 4, 1);
        pool3d_kernel<<<launch1(36L * 16 * 16 * 16), 256, 0, stream>>>(A, XA, 32, 2, 1, 0);
    }
    auto sm_layer = [&](int pb, const float* xin, int Din, float* xout) {
        int S = Din * Din * Din;
        conv36(P(pb + 0), xin, Q,  S);
        conv36(P(pb + 1), xin, Kb, S);
        conv36(P(pb + 2), xin, Vb, S);
        build_rel3d_kernel<<<launch1(36 * 27), 256, 0, stream>>>(P(pb + 3), P(pb + 4), P(pb + 5), REL, 3);
        attn3d_kernel<<<launch1(36L * S), 256, 0, stream>>>(Q, Kb, Vb, REL, A, Din, Din, 3, 1);
        pool3d_kernel<<<launch1(36L * S / 8), 256, 0, stream>>>(A, xout, Din, 2, 1, 0);
    };
    sm_layer(8,  XA, 16, XB);   // -> [36,8^3]
    sm_layer(14, XB, 8,  XC);   // -> [36,4^3]
    channel_attn_kernel<<<1, 256, 0, stream>>>(XC, P(32), P(33), P(34), P(35), 64);
    chan_meanmax_kernel<<<1, 256, 0, stream>>>(XC, MM, 64);
    sattn3d_kernel<<<1, 256, 0, stream>>>(XC, MM, P(40), P(41), 4);
    pool3d_kernel<<<launch1(36L * 64), 256, 0, stream>>>(XA, XC, 16, 4, 0, 1); // + pool4(skip)
    sm_layer(20, XC, 4, XB);    // -> [36,2^3]
    sm_layer(26, XB, 2, XD);    // -> [36,1]
    channel_attn_kernel<<<1, 256, 0, stream>>>(XD, P(36), P(37), P(38), P(39), 1);
    chan_meanmax_kernel<<<1, 256, 0, stream>>>(XD, MM, 1);
    sattn3d_kernel<<<1, 256, 0, stream>>>(XD, MM, P(42), P(43), 1);
    pool3d_kernel<<<launch1(36L), 256, 0, stream>>>(XC, XD, 4, 4, 0, 1);       // + pool4(skip)
    linear(P(44), XD, P(45), F1, 512, 36, 1);                                   // li1 + relu
    linear(P(46), F1, P(47), Z, 256, 512, 1);                                   // li2 + relu -> Z[0:256]

    // =========================== TS branch (1D) ===========================
    {   // c1: Cin=1, Lin=2049, k=4, pad=1 -> Lout=2048, pool2 -> XA [36,1024]
        int Lin = 2049, Lout = 2048;
        conv1(P(48), P(1), Q,  Lin);
        conv1(P(49), P(1), Kb, Lin);
        conv1(P(50), P(1), Vb, Lin);
        attn1d_kernel<<<launch1(36L * Lout), 256, 0, stream>>>(Q, Kb, Vb, P(51), A, Lin, Lout, 4, 1);
        pool1d_kernel<<<launch1(36L * 1024), 256, 0, stream>>>(A, XA, 2048, 2, 1, 0);
    }
    auto ts_layer = [&](int pb, const float* xin, int L, float* xout) {
        conv36(P(pb + 0), xin, Q,  L);
        conv36(P(pb + 1), xin, Kb, L);
        conv36(P(pb + 2), xin, Vb, L);
        attn1d_kernel<<<launch1(36L * L), 256, 0, stream>>>(Q, Kb, Vb, P(pb + 3), A, L, L, 3, 1);
        pool1d_kernel<<<launch1(36L * L / 2), 256, 0, stream>>>(A, xout, L, 2, 1, 0);
    };
    ts_layer(52, XA, 1024, XB);  // -> [36,512]
    ts_layer(56, XB, 512,  XC);  // -> [36,256]
    channel_attn_kernel<<<1, 256, 0, stream>>>(XC, P(68), P(69), P(70), P(71), 256);
    pool1d_kernel<<<launch1(36L * 256), 256, 0, stream>>>(XA, XC, 1024, 4, 0, 1); // + pool4(skip)
    ts_layer(60, XC, 256, XB);   // -> [36,128]
    ts_layer(64, XB, 128, XD);   // -> [36,64]
    channel_attn_kernel<<<1, 256, 0, stream>>>(XD, P(72), P(73), P(74), P(75), 64);
    pool1d_kernel<<<launch1(36L * 64), 256, 0, stream>>>(XC, XD, 256, 4, 0, 1);   // + pool4(skip)
    linear(P(76), XD, P(77), F1, 1152, 2304, 1);                                  // li1 + relu
    linear(P(78), F1, P(79), Z + 256, 576, 1152, 1);                              // li2 -> Z[256:832]

    // =============================== Head ===============================
    linear(P(80), Z,  P(81), H4, 416, 832, 1);
    linear(P(82), H4, P(83), H5, 208, 416, 1);
    linear(P(84), H5, P(85), (float*)d_out, 2, 208, 0);
}